// GCNLayer_7834020348104
// MI455X (gfx1250) — compile-verified
//
#include <hip/hip_runtime.h>

typedef __attribute__((ext_vector_type(2))) float v2f;
typedef __attribute__((ext_vector_type(8))) float v8f;

#define D_FEAT 64
#define UNITS 64
#define ROWS_PER_BLOCK 16
#define THREADS_PER_BLOCK 128   // 4 wave32s

// Fused GCN layer: segment-sum aggregation (sorted dst) + f32 WMMA projection.
__global__ __launch_bounds__(THREADS_PER_BLOCK)
void gcn_fused_kernel(const float* __restrict__ nodes,      // [N, 64]
                      const float* __restrict__ edge_vals,  // [E]
                      const float* __restrict__ kern,       // [64, 64] row-major
                      const int*   __restrict__ src_idx,    // [E]
                      const int*   __restrict__ dst_idx,    // [E], sorted
                      float*       __restrict__ out,        // [N, 64]
                      int n_edges)
{
    __shared__ float agg[ROWS_PER_BLOCK][D_FEAT];   // 4 KB tile of aggregated rows

    const int tid  = threadIdx.x;
    const int lane = tid & 31;
    const int wave = tid >> 5;                       // 0..3
    const int rowBase = blockIdx.x * ROWS_PER_BLOCK;

    // ---------------- Phase 1: aggregation ----------------
    // Each wave owns 4 node rows. dst_idx is sorted, so each node's edges are
    // a contiguous range found by binary search -> no atomics, deterministic.
    for (int r = wave; r < ROWS_PER_BLOCK; r += (THREADS_PER_BLOCK / 32)) {
        const int node = rowBase + r;

        int lo = 0, hi = n_edges;
        while (lo < hi) {                            // lower_bound(node)
            int mid = (lo + hi) >> 1;
            if (dst_idx[mid] < node) lo = mid + 1; else hi = mid;
        }
        const int start = lo;
        hi = n_edges;                                // lower_bound(node + 1)
        while (lo < hi) {
            int mid = (lo + hi) >> 1;
            if (dst_idx[mid] < node + 1) lo = mid + 1; else hi = mid;
        }
        const int end = lo;

        // Lane holds features [2*lane, 2*lane+1]; wave reads 256B contiguous per edge.
        v2f acc; acc.x = 0.0f; acc.y = 0.0f;
        const int f = lane * 2;
        for (int e = start; e < end; ++e) {
            const float w = edge_vals[e];            // wave-uniform broadcast load
            const int   s = src_idx[e];              // wave-uniform
            const v2f   x = *(const v2f*)(nodes + (size_t)s * D_FEAT + f);
            acc.x += w * x.x;
            acc.y += w * x.y;
        }
        *(v2f*)(&agg[r][f]) = acc;                   // ds_store_b64
    }
    __syncthreads();

    // ---------------- Phase 2: out = agg(16x64) @ kern(64x64) ----------------
    // One 16-wide N-tile per wave; accumulate K=64 with 16x v_wmma_f32_16x16x4_f32.
    // ISA 7.12.2 layouts:
    //   A 16x4 (f32): lanes 0-15 -> M=lane,   K={k0,k0+1}
    //                 lanes16-31 -> M=lane-16,K={k0+2,k0+3}
    //   B 4x16 mirrors A over N.
    //   C/D 16x16:    VGPR r, lanes 0-15 -> (M=r,   N=lane)
    //                         lanes16-31 -> (M=r+8, N=lane-16)
    const int n0   = wave * 16;
    const int half = lane >> 4;                      // 0 or 1
    const int l16  = lane & 15;
    const int kOfs = half * 2;

    v8f c = {};
    #pragma unroll
    for (int k0 = 0; k0 < D_FEAT; k0 += 4) {
        v2f a;
        a.x = agg[l16][k0 + kOfs];
        a.y = agg[l16][k0 + kOfs + 1];
        v2f b;
        b.x = kern[(size_t)(k0 + kOfs)     * UNITS + n0 + l16];
        b.y = kern[(size_t)(k0 + kOfs + 1) * UNITS + n0 + l16];
        // 8 args: (neg_a, A, neg_b, B, c_mod, C, reuse_a, reuse_b)
        c = __builtin_amdgcn_wmma_f32_16x16x4_f32(false, a, false, b,
                                                  (short)0, c, false, false);
    }

    const int outRow0 = rowBase + half * 8;
    const int outCol  = n0 + l16;
    #pragma unroll
    for (int r = 0; r < 8; ++r) {
        out[(size_t)(outRow0 + r) * UNITS + outCol] = c[r];
    }
}

extern "C" void kernel_launch(void* const* d_in, const int* in_sizes, int n_in,
                              void* d_out, int out_size, void* d_ws, size_t ws_size,
                              hipStream_t stream) {
    (void)n_in; (void)d_ws; (void)ws_size; (void)out_size;

    const float* nodes     = (const float*)d_in[0];
    const float* edge_vals = (const float*)d_in[1];
    const float* kern      = (const float*)d_in[2];
    const int*   src_idx   = (const int*)d_in[3];
    const int*   dst_idx   = (const int*)d_in[4];
    float*       out       = (float*)d_out;

    const int n_nodes = in_sizes[0] / D_FEAT;        // 65536
    const int n_edges = in_sizes[1];                 // 1048576

    const int grid = n_nodes / ROWS_PER_BLOCK;       // 4096 blocks
    gcn_fused_kernel<<<grid, THREADS_PER_BLOCK, 0, stream>>>(
        nodes, edge_vals, kern, src_idx, dst_idx, out, n_edges);
}